// MixHopLayer_4337916969700
// MI455X (gfx1250) — compile-verified
//
#include <hip/hip_runtime.h>

// ---------------------------------------------------------------------------
// MixHop layer for MI455X (gfx1250, wave32).
//   out[b, p*64+o, n, t] = leaky_relu( (A^p (X Wp + bp))[b,o,n,t] ),  p=0,1,2
// Dominant cost: adj@H GEMMs (155 GFLOP) -> f32 WMMA 16x16x4, fed by
// double-buffered GLOBAL_LOAD_ASYNC_TO_LDS_B128 staging (ASYNCcnt).
// ---------------------------------------------------------------------------

typedef float v2f __attribute__((ext_vector_type(2)));
typedef float v8f __attribute__((ext_vector_type(8)));
typedef int   vs4i __attribute__((__vector_size__(16)));   // matches builtin param

#define N_NODES 2048
#define T_STEPS 12
#define F_IN    64
#define F_OUT   64
#define RJ      768                 // F_OUT * T_STEPS; H column index j = o*12 + t
#define RROWS   24576               // N_NODES * T_STEPS (flat row r = n*12 + t)
#define XBATCH  (F_IN * RROWS)      // elements of x per batch
#define NBATCH  8

// LDS tile strides: multiples of 4 floats (16 B) so async B128 stores are
// naturally aligned; 36 mod 64 has cycle length 16 -> the 16-lane A-fragment
// column reads still land in 16 distinct banks.
#define LDA 36
#define LDB 68

union V8 { v8f v; float f[8]; };

__device__ __forceinline__ float lrelu(float v) { return v > 0.0f ? v : 0.01f * v; }

__device__ __forceinline__ v8f wmma4(v2f a, v2f b, v8f c) {
  // D = A(16x4 f32) * B(4x16 f32) + C(16x16 f32)
  return __builtin_amdgcn_wmma_f32_16x16x4_f32(false, a, false, b, (short)0, c,
                                               false, false);
}

// ---- async global -> LDS helpers ------------------------------------------
__device__ __forceinline__ void async_copy16(const float* g, float* l) {
#if __has_builtin(__builtin_amdgcn_global_load_async_to_lds_b128)
  __builtin_amdgcn_global_load_async_to_lds_b128(
      (__attribute__((address_space(1))) vs4i*)g,
      (__attribute__((address_space(3))) vs4i*)l, 0, 0);
#else
  unsigned int lofs = (unsigned int)(unsigned long long)l;  // LDS offset = low 32 bits
  unsigned long long ga = (unsigned long long)g;
  asm volatile("global_load_async_to_lds_b128 %0, %1, off"
               :: "v"(lofs), "v"(ga) : "memory");
#endif
}

__device__ __forceinline__ void async_wait0() {
#if __has_builtin(__builtin_amdgcn_s_wait_asynccnt)
  __builtin_amdgcn_s_wait_asynccnt(0);
#else
  asm volatile("s_wait_asynccnt 0x0" ::: "memory");
#endif
}

// ---------------------------------------------------------------------------
// Kernel 1: feature transform.  Rows = flat (n,t), K = F_IN, cols = 3*F_OUT.
// WG tile: 128 rows x 192 cols; 8 waves, each owns 32 rows and 3x(2x16) cols.
// x tile transposed into LDS (pad 66 -> conflict-free v2f fragment reads).
// ---------------------------------------------------------------------------
__global__ __launch_bounds__(256) void feat_wmma(
    const float* __restrict__ x,
    const float* __restrict__ W0, const float* __restrict__ b0,
    const float* __restrict__ W1, const float* __restrict__ b1,
    const float* __restrict__ W2, const float* __restrict__ b2,
    float* __restrict__ out, float* __restrict__ y1, float* __restrict__ y2)
{
  __shared__ float sX[128 * 66];

  const int tid   = threadIdx.x;
  const int lane  = tid & 31;
  const int wave  = tid >> 5;
  const int batch = blockIdx.y;
  const int row0  = blockIdx.x * 128;

  // Transposed load: for fixed f, rows are contiguous in memory -> coalesced.
  const float* xb = x + (size_t)batch * XBATCH + row0;
  for (int i = tid; i < 128 * 64; i += 256) {
    int f = i >> 7;
    int r = i & 127;
    sX[r * 66 + f] = xb[(size_t)f * RROWS + r];
  }
  __syncthreads();

  const int lm  = lane & 15;
  const int lh  = lane >> 4;
  const int rb  = (wave & 3) * 32;   // wave row offset in tile
  const int cb0 = (wave >> 2) * 32;  // wave col offset within one power's 64 cols

  const float* Wp[3] = {W0, W1, W2};
  const float* bp[3] = {b0, b1, b2};

  v8f acc[3][2][2];
#pragma unroll
  for (int p = 0; p < 3; ++p)
#pragma unroll
    for (int ih = 0; ih < 2; ++ih)
#pragma unroll
      for (int jh = 0; jh < 2; ++jh) acc[p][ih][jh] = (v8f){};

#pragma unroll
  for (int kk = 0; kk < F_IN; kk += 4) {
    const int kb = kk + lh * 2;
    v2f a0 = *(const v2f*)&sX[(rb + lm) * 66 + kb];
    v2f a1 = *(const v2f*)&sX[(rb + 16 + lm) * 66 + kb];
#pragma unroll
    for (int p = 0; p < 3; ++p) {
      const float* W = Wp[p];
#pragma unroll
      for (int jh = 0; jh < 2; ++jh) {
        const int o = cb0 + jh * 16 + lm;
        v2f bf;
        bf.x = W[kb * F_OUT + o];        // tiny, L0/L2-hot, coalesced per half-wave
        bf.y = W[(kb + 1) * F_OUT + o];
        acc[p][0][jh] = wmma4(a0, bf, acc[p][0][jh]);
        acc[p][1][jh] = wmma4(a1, bf, acc[p][1][jh]);
      }
    }
  }

  // Epilogue: p0 -> relu to out; p1/p2 -> (+bias) raw to workspace H[n][o*12+t].
#pragma unroll
  for (int p = 0; p < 3; ++p) {
#pragma unroll
    for (int ih = 0; ih < 2; ++ih) {
#pragma unroll
      for (int jh = 0; jh < 2; ++jh) {
        const int o      = cb0 + jh * 16 + lm;
        const float bias = bp[p][o];
        const int rbase  = row0 + rb + ih * 16 + lh * 8;  // flat row = n*12 + t
        V8 u; u.v = acc[p][ih][jh];
        if (p == 0) {
          float* dst = out + ((size_t)batch * 192 + o) * RROWS;
#pragma unroll
          for (int i = 0; i < 8; ++i) dst[rbase + i] = lrelu(u.f[i] + bias);
        } else {
          float* yb = (p == 1 ? y1 : y2) + (size_t)batch * N_NODES * RJ;
#pragma unroll
          for (int i = 0; i < 8; ++i) {
            const int r = rbase + i;
            const int n = r / T_STEPS;
            const int t = r - n * T_STEPS;
            yb[(size_t)n * RJ + o * T_STEPS + t] = u.f[i] + bias;
          }
        }
      }
    }
  }
}

// ---------------------------------------------------------------------------
// Kernel 2: diffusion GEMM  C[2048 x 768] = adj[2048 x 2048] @ H[2048 x 768].
// WG tile 128x64, 8 waves (4x2), 32x32 per wave (4 x v8f accumulators).
// K staged 32 at a time into double-buffered LDS via async B128 copies:
// prefetch(next) overlaps WMMA(cur); per-wave s_wait_asynccnt 0 + s_barrier
// guarantees the whole tile is resident before anyone reads it.
// ---------------------------------------------------------------------------
__device__ __forceinline__ void stage_tiles_async(
    const float* __restrict__ A, const float* __restrict__ Hb,
    float* sAbuf, float* sBbuf, int row0, int col0, int k0, int tid)
{
  // A tile 128x32 (16 KB): 1024 float4s, 4 per thread.
#pragma unroll
  for (int ii = 0; ii < 4; ++ii) {
    const int i = tid + ii * 256;
    const int r = i >> 3;
    const int c = (i & 7) * 4;
    async_copy16(&A[(size_t)(row0 + r) * N_NODES + k0 + c], &sAbuf[r * LDA + c]);
  }
  // B tile 32x64 (8 KB): 512 float4s, 2 per thread.
#pragma unroll
  for (int ii = 0; ii < 2; ++ii) {
    const int i = tid + ii * 256;
    const int r = i >> 4;
    const int c = (i & 15) * 4;
    async_copy16(&Hb[(size_t)(k0 + r) * RJ + col0 + c], &sBbuf[r * LDB + c]);
  }
}

__global__ __launch_bounds__(256) void diffuse_wmma(
    const float* __restrict__ adj, const float* __restrict__ H,
    float* __restrict__ dst, int apply_relu, int slot)
{
  __shared__ float sA[2][128 * LDA];   // 2 x 18 KB
  __shared__ float sB[2][32 * LDB];    // 2 x 8.5 KB

  const int tid   = threadIdx.x;
  const int lane  = tid & 31;
  const int wave  = tid >> 5;
  const int batch = blockIdx.z;
  const int row0  = blockIdx.y * 128;
  const int col0  = blockIdx.x * 64;

  const float* A  = adj + (size_t)batch * N_NODES * N_NODES;
  const float* Hb = H   + (size_t)batch * N_NODES * RJ;

  const int lm = lane & 15;
  const int lh = lane >> 4;
  const int rb = (wave & 3) * 32;
  const int cb = (wave >> 2) * 32;

  v8f acc00 = (v8f){}, acc01 = (v8f){}, acc10 = (v8f){}, acc11 = (v8f){};

  // Prologue: stage K-chunk 0 into buffer 0.
  stage_tiles_async(A, Hb, &sA[0][0], &sB[0][0], row0, col0, 0, tid);
  async_wait0();
  __syncthreads();

  for (int k0 = 0; k0 < N_NODES; k0 += 32) {
    const int cur = (k0 >> 5) & 1;
    const int nxt = cur ^ 1;

    // Prefetch next K-chunk while this one computes. Safe to overwrite nxt:
    // the barrier at the end of the previous iteration guarantees all waves
    // finished reading it.
    if (k0 + 32 < N_NODES)
      stage_tiles_async(A, Hb, &sA[nxt][0], &sB[nxt][0], row0, col0, k0 + 32, tid);

    const float* cA = &sA[cur][0];
    const float* cB = &sB[cur][0];
#pragma unroll
    for (int kk = 0; kk < 32; kk += 4) {
      const int kb = kk + lh * 2;
      v2f a0 = *(const v2f*)&cA[(rb + lm) * LDA + kb];
      v2f a1 = *(const v2f*)&cA[(rb + 16 + lm) * LDA + kb];
      v2f bf0, bf1;
      bf0.x = cB[kb * LDB + cb + lm];
      bf0.y = cB[(kb + 1) * LDB + cb + lm];
      bf1.x = cB[kb * LDB + cb + 16 + lm];
      bf1.y = cB[(kb + 1) * LDB + cb + 16 + lm];
      acc00 = wmma4(a0, bf0, acc00);
      acc01 = wmma4(a0, bf1, acc01);
      acc10 = wmma4(a1, bf0, acc10);
      acc11 = wmma4(a1, bf1, acc11);
    }

    // Drain own async copies, then rendezvous: after the barrier, every
    // wave's slice of the next tile is resident in LDS.
    async_wait0();
    __syncthreads();
  }

  // Epilogue
  const int gr = row0 + rb;
  const int gc = col0 + cb;
#pragma unroll
  for (int ih = 0; ih < 2; ++ih) {
#pragma unroll
    for (int jh = 0; jh < 2; ++jh) {
      V8 u;
      u.v = (ih == 0) ? (jh == 0 ? acc00 : acc01) : (jh == 0 ? acc10 : acc11);
      const int col   = gc + jh * 16 + lm;       // j = o*12 + t
      const int rbase = gr + ih * 16 + lh * 8;   // node n
      if (apply_relu) {
        const int o = col / T_STEPS;
        const int t = col - o * T_STEPS;
        float* op = dst + (((size_t)batch * 192 + slot * 64 + o) * N_NODES) * T_STEPS + t;
#pragma unroll
        for (int i = 0; i < 8; ++i)
          op[(size_t)(rbase + i) * T_STEPS] = lrelu(u.f[i]);
      } else {
        float* op = dst + (size_t)batch * N_NODES * RJ + col;
#pragma unroll
        for (int i = 0; i < 8; ++i)
          op[(size_t)(rbase + i) * RJ] = u.f[i];
      }
    }
  }
}

// ---------------------------------------------------------------------------
extern "C" void kernel_launch(void* const* d_in, const int* in_sizes, int n_in,
                              void* d_out, int out_size, void* d_ws, size_t ws_size,
                              hipStream_t stream) {
  (void)in_sizes; (void)n_in; (void)out_size; (void)ws_size;

  const float* x   = (const float*)d_in[0];
  const float* adj = (const float*)d_in[1];
  const float* W0  = (const float*)d_in[2];
  const float* b0  = (const float*)d_in[3];
  const float* W1  = (const float*)d_in[4];
  const float* b1  = (const float*)d_in[5];
  const float* W2  = (const float*)d_in[6];
  const float* b2  = (const float*)d_in[7];
  float* out = (float*)d_out;

  // Double-buffered workspace: 2 x [B, N, 768] f32 = 100.7 MB.
  const size_t bufElems = (size_t)NBATCH * N_NODES * RJ;
  float* buf0 = (float*)d_ws;
  float* buf1 = buf0 + bufElems;

  dim3 blk(256);

  // 1) Y0 -> out (relu);  Y1 -> buf0;  Y2 -> buf1   (bias included)
  feat_wmma<<<dim3(RROWS / 128, NBATCH), blk, 0, stream>>>(
      x, W0, b0, W1, b1, W2, b2, out, buf0, buf1);

  dim3 ggrid(RJ / 64, N_NODES / 128, NBATCH);
  // 2) out slot1 = relu(adj @ Y1)
  diffuse_wmma<<<ggrid, blk, 0, stream>>>(adj, buf0, out, 1, 1);
  // 3) buf0 = adj @ Y2            (raw; stream-ordered after (2) reads buf0)
  diffuse_wmma<<<ggrid, blk, 0, stream>>>(adj, buf1, buf0, 0, 0);
  // 4) out slot2 = relu(adj @ buf0)
  diffuse_wmma<<<ggrid, blk, 0, stream>>>(adj, buf0, out, 1, 2);
}